// MambaBackbone_44710609551995
// MI455X (gfx1250) — compile-verified
//
#include <hip/hip_runtime.h>
#include <hip/hip_bf16.h>

// ---------------------------------------------------------------------------
// Mamba backbone on gfx1250 (MI455X), fp32 end-to-end.
//  - GEMMs: V_WMMA_F32_16X16X4_F32 (native fp32 WMMA, matches reference
//    numerics). One wave per 16x16 tile; A and B both load as contiguous
//    float2 per lane per k-step (ISA 7.12.2 f32 layouts).
//  - Selective scan: time-tiled (T=8), double-buffered LDS, with the shared
//    B/C block prefetched by the Tensor Data Mover (tensor_load_to_lds +
//    s_wait_tensorcnt) one tile ahead of the compute.
// ---------------------------------------------------------------------------

typedef float v2f __attribute__((ext_vector_type(2)));
typedef float v8f __attribute__((ext_vector_type(8)));
typedef unsigned int v4u __attribute__((ext_vector_type(4)));
typedef int v8i __attribute__((ext_vector_type(8)));
typedef int v4i __attribute__((ext_vector_type(4)));

namespace mamba {
constexpr int B       = 2;
constexpr int L       = 1024;
constexpr int DM      = 768;    // d_model
constexpr int DI      = 1536;   // d_inner
constexpr int DS      = 16;     // d_state
constexpr int DCONV   = 4;
constexpr int DTRANK  = 48;
constexpr int XDBL    = DTRANK + 2 * DS;   // 80
constexpr int ROWS    = B * L;             // 2048
constexpr int NLAYER  = 4;
constexpr int ST      = 8;                 // scan time-tile
}

// ---------------------------------------------------------------------------
// Fused residual add + LayerNorm(768). One 256-thread block per row.
// ---------------------------------------------------------------------------
__global__ __launch_bounds__(256) void mamba_add_ln(
    const float* __restrict__ h, const float* __restrict__ rin,
    float* __restrict__ rout, float* __restrict__ nout,
    const float* __restrict__ w, const float* __restrict__ b) {
  using namespace mamba;
  __shared__ float red[2][8];
  const int row = blockIdx.x;
  const int tid = threadIdx.x;
  const float* hp = h + (size_t)row * DM;
  const float* rp = rin ? rin + (size_t)row * DM : nullptr;

  float v[3];
  float s = 0.f, s2 = 0.f;
#pragma unroll
  for (int i = 0; i < 3; ++i) {
    const int c = tid + i * 256;
    float x = hp[c] + (rp ? rp[c] : 0.f);
    v[i] = x;
    s += x;
    s2 += x * x;
  }
#pragma unroll
  for (int off = 16; off > 0; off >>= 1) {
    s  += __shfl_down(s,  off, 32);
    s2 += __shfl_down(s2, off, 32);
  }
  const int wv = tid >> 5, ln = tid & 31;
  if (ln == 0) { red[0][wv] = s; red[1][wv] = s2; }
  __syncthreads();
  if (wv == 0) {
    s  = (ln < 8) ? red[0][ln] : 0.f;
    s2 = (ln < 8) ? red[1][ln] : 0.f;
#pragma unroll
    for (int off = 4; off > 0; off >>= 1) {
      s  += __shfl_down(s,  off, 32);
      s2 += __shfl_down(s2, off, 32);
    }
    if (ln == 0) { red[0][0] = s; red[1][0] = s2; }
  }
  __syncthreads();
  const float mean = red[0][0] * (1.f / DM);
  const float var  = red[1][0] * (1.f / DM) - mean * mean;
  const float inv  = rsqrtf(var + 1e-5f);
  float* rop = rout + (size_t)row * DM;
  float* nop = nout + (size_t)row * DM;
#pragma unroll
  for (int i = 0; i < 3; ++i) {
    const int c = tid + i * 256;
    rop[c] = v[i];
    nop[c] = (v[i] - mean) * inv * w[c] + b[c];
  }
}

// ---------------------------------------------------------------------------
// WMMA fp32 GEMM: C[M,N] = X[M,K(lda)] * W[N,K(ldb)]^T  (+ epilogue)
// EPI 0: plain store. EPI 1: softplus(acc + bias[n]).
// ---------------------------------------------------------------------------
template <int EPI>
__global__ __launch_bounds__(256) void mamba_gemm_wmma(
    const float* __restrict__ X, int lda,
    const float* __restrict__ W, int ldb,
    const float* __restrict__ bias,
    float* __restrict__ C,
    int M, int N, int K) {
  const int lane = threadIdx.x & 31;
  const int wave = (blockIdx.x * blockDim.x + threadIdx.x) >> 5;
  const int tilesN = N >> 4;
  const int tilesTotal = (M >> 4) * tilesN;
  if (wave >= tilesTotal) return;   // whole-wave exit: EXEC stays all-ones
  const int tm = wave / tilesN;
  const int tn = wave % tilesN;
  const int lr = lane & 15;
  const int hi = lane >> 4;

  const float* xrow = X + (size_t)(tm * 16 + lr) * lda + 2 * hi;
  const float* wrow = W + (size_t)(tn * 16 + lr) * ldb + 2 * hi;

  v8f acc = {0.f, 0.f, 0.f, 0.f, 0.f, 0.f, 0.f, 0.f};
  for (int k = 0; k < K; k += 4) {
    v2f a = *(const v2f*)(xrow + k);
    v2f bb = *(const v2f*)(wrow + k);
    acc = __builtin_amdgcn_wmma_f32_16x16x4_f32(
        /*neg_a=*/false, a, /*neg_b=*/false, bb,
        /*c_mod=*/(short)0, acc, /*reuse_a=*/false, /*reuse_b=*/false);
  }

  const int n  = tn * 16 + lr;
  const int m0 = tm * 16 + 8 * hi;
#pragma unroll
  for (int r = 0; r < 8; ++r) {
    float v = acc[r];
    if (EPI == 1) {            // softplus(acc + dt_bias)
      v += bias[n];
      v = (v > 20.f) ? v : log1pf(__expf(v));
    }
    C[(size_t)(m0 + r) * N + n] = v;
  }
}

// ---------------------------------------------------------------------------
// Causal depthwise conv1d (k=4) + SiLU. Reads x half of xz (row stride 2*DI).
// ---------------------------------------------------------------------------
__global__ __launch_bounds__(256) void mamba_conv_silu(
    const float* __restrict__ xz, const float* __restrict__ cw,
    const float* __restrict__ cb, float* __restrict__ out) {
  using namespace mamba;
  const int idx = blockIdx.x * blockDim.x + threadIdx.x;
  if (idx >= B * L * DI) return;
  const int d  = idx % DI;
  const int l  = (idx / DI) % L;
  const int bb = idx / (DI * L);
  float acc = cb[d];
#pragma unroll
  for (int kk = 0; kk < DCONV; ++kk) {
    const int ls = l + kk - (DCONV - 1);
    if (ls >= 0)
      acc += cw[d * DCONV + kk] * xz[(size_t)(bb * L + ls) * (2 * DI) + d];
  }
  out[idx] = acc / (1.f + __expf(-acc));   // SiLU
}

// ---------------------------------------------------------------------------
// Selective scan, time-tiled by ST=8 with TDM prefetch of the shared B/C
// block.  One thread per (batch, channel); 16-element state in registers.
// Each tile's B/C rows (32 floats every 80-float row => strided 2D tile) are
// DMA'd into LDS by the Tensor Data Mover one tile ahead (double-buffered).
// Fuses D-skip and SiLU(z) gating.
// ---------------------------------------------------------------------------
__global__ __launch_bounds__(256) void mamba_scan(
    const float* __restrict__ dtb,   // (ROWS, DI) softplus'd dt
    const float* __restrict__ xc,    // (ROWS, DI) conv+silu x
    const float* __restrict__ xdbl,  // (ROWS, 80) [dt_lowrank | B | C]
    const float* __restrict__ xz,    // (ROWS, 2*DI) (z = second half)
    const float* __restrict__ A_log, // (DI, DS)
    const float* __restrict__ Dsk,   // (DI)
    float* __restrict__ y) {         // (ROWS, DI)
  using namespace mamba;
  __shared__ float sBC[2][ST * 2 * DS];    // double buffer, 1 KB each
  const int tid = threadIdx.x;
  const int gid = blockIdx.x * 256 + tid;  // 0 .. B*DI-1 (grid = 12 blocks)
  const int d   = gid % DI;
  const int bb  = gid / DI;                // uniform per block (256 | 1536)

  // LDS byte offset of the staging buffer (low 32 bits of the generic
  // address of a __shared__ object == AS3 offset on gfx1250).
  const unsigned ldsBase =
      (unsigned)(uintptr_t)(const void*)(&sBC[0][0]);
  const float* gBase = xdbl + (size_t)bb * L * XDBL + DTRANK; // col 48 onward

  // Issue a TDM load of tile `t`'s B/C block (32 x ST fp32, row stride 80)
  // into LDS buffer `pb`.  All D# fields per CDNA5 ISA sec. 8.3/8.4.
  auto tdm_issue = [&](int t, int pb) {
    const unsigned long long ga =
        (unsigned long long)(uintptr_t)(gBase + (size_t)t * ST * XDBL);
    v4u g0;
    g0[0] = 1u;                                     // count=1, no gather
    g0[1] = ldsBase + (unsigned)pb * (ST * 2 * DS * 4);  // lds_addr (bytes)
    g0[2] = (unsigned)ga;                           // global_addr[31:0]
    g0[3] = (unsigned)((ga >> 32) & 0x01FFFFFFu)    // global_addr[56:32]
            | (2u << 30);                           // type=2 ("image")
    v8i g1;
    g1[0] = (int)(2u << 16);          // data_size=2 (4 bytes), mask=0
    g1[1] = (int)((2u * DS) << 16);   // tensor_dim0 = 32 (low 16 in [31:16])
    g1[2] = (int)((unsigned)ST << 16);// tensor_dim1 = ST
    g1[3] = (int)((2u * DS) << 16);   // tile_dim0 = 32  (bits 127:112)
    g1[4] = ST;                       // tile_dim1 = ST  (bits 143:128)
    g1[5] = XDBL;                     // tensor_dim0_stride = 80
    g1[6] = 0;
    g1[7] = 0;
    v4i g2 = {0, 0, 0, 0};
    v4i g3 = {0, 0, 0, 0};
    v8i g4 = {0, 0, 0, 0, 0, 0, 0, 0};   // 6-arg toolchain form: extra group
    __builtin_amdgcn_tensor_load_to_lds(g0, g1, g2, g3, g4, 0);
  };

  float A[DS], h[DS];
#pragma unroll
  for (int s = 0; s < DS; ++s) {
    A[s] = -__expf(A_log[d * DS + s]);
    h[s] = 0.f;
  }
  const float Dv = Dsk[d];

  if (tid < 32) tdm_issue(0, 0);     // prologue: prefetch tile 0 (wave 0)

  constexpr int NT = L / ST;         // 128 tiles
  for (int t = 0; t < NT; ++t) {
    const int pb = t & 1;
    if (tid < 32) __builtin_amdgcn_s_wait_tensorcnt(0);  // tile t landed
    __syncthreads();                                     // visible to all
    // Prefetch tile t+1 into the other buffer.  Safe: the issue of tile
    // t+2 (which overwrites buffer pb) happens only after the top barrier
    // of tile t+1, which follows every wave's reads of tile t.
    if (tid < 32 && t + 1 < NT) tdm_issue(t + 1, pb ^ 1);

    // Batch-load this thread's dt/x/z for all ST steps (one load clause).
    const size_t r0 = (size_t)(bb * L + t * ST);
    float dtv[ST], xv[ST], zv[ST];
#pragma unroll
    for (int j = 0; j < ST; ++j) {
      dtv[j] = dtb[(r0 + j) * DI + d];
      xv[j]  = xc[(r0 + j) * DI + d];
      zv[j]  = xz[(r0 + j) * (2 * DI) + DI + d];
    }

#pragma unroll
    for (int j = 0; j < ST; ++j) {
      const float dt  = dtv[j];
      const float x   = xv[j];
      const float z   = zv[j];
      const float dtx = dt * x;
      const float* bc = &sBC[pb][j * 2 * DS];
      float acc = 0.f;
#pragma unroll
      for (int s = 0; s < DS; ++s) {
        const float dA = __expf(dt * A[s]);
        h[s] = dA * h[s] + dtx * bc[s];
        acc += h[s] * bc[DS + s];
      }
      acc += x * Dv;
      acc *= z / (1.f + __expf(-z));   // gate with SiLU(z)
      y[(r0 + j) * DI + d] = acc;
    }
  }
}

// ---------------------------------------------------------------------------
// Host-side orchestration
// ---------------------------------------------------------------------------
static inline int gemm_grid(int M, int N) {
  const int tiles = (M / 16) * (N / 16);
  return (tiles + 7) / 8;            // 8 waves (16x16 tiles) per 256-thr block
}

extern "C" void kernel_launch(void* const* d_in, const int* in_sizes, int n_in,
                              void* d_out, int out_size, void* d_ws, size_t ws_size,
                              hipStream_t stream) {
  using namespace mamba;
  const float* emb       = (const float*)d_in[0];
  const float* norm_w    = (const float*)d_in[1];
  const float* norm_b    = (const float*)d_in[2];
  const float* in_proj_w = (const float*)d_in[3];
  const float* conv_w    = (const float*)d_in[4];
  const float* conv_b    = (const float*)d_in[5];
  const float* x_proj_w  = (const float*)d_in[6];
  const float* dt_proj_w = (const float*)d_in[7];
  const float* dt_proj_b = (const float*)d_in[8];
  const float* A_log     = (const float*)d_in[9];
  const float* D_skip    = (const float*)d_in[10];
  const float* out_proj_w= (const float*)d_in[11];
  const float* norm_f_w  = (const float*)d_in[12];
  const float* norm_f_b  = (const float*)d_in[13];

  // workspace carve-up (floats)
  float* ws = (float*)d_ws;
  float* residual = ws;                     ws += (size_t)ROWS * DM;
  float* normed   = ws;                     ws += (size_t)ROWS * DM;
  float* hidden   = ws;                     ws += (size_t)ROWS * DM;
  float* xz       = ws;                     ws += (size_t)ROWS * 2 * DI;
  float* xconv    = ws;                     ws += (size_t)ROWS * DI;
  float* xdbl     = ws;                     ws += (size_t)ROWS * XDBL;
  float* dtbuf    = ws;                     ws += (size_t)ROWS * DI;
  float* ybuf     = ws;                     ws += (size_t)ROWS * DI;
  (void)ws_size; (void)in_sizes; (void)n_in; (void)out_size;

  const int convN = B * L * DI;

  for (int layer = 0; layer < NLAYER; ++layer) {
    const float* hid = (layer == 0) ? emb : hidden;
    const float* rin = (layer == 0) ? nullptr : residual;

    // residual add + LayerNorm
    mamba_add_ln<<<ROWS, 256, 0, stream>>>(
        hid, rin, residual, normed,
        norm_w + (size_t)layer * DM, norm_b + (size_t)layer * DM);

    // in_proj: (ROWS, 2*DI) = normed (ROWS,DM) x W(2*DI, DM)^T
    mamba_gemm_wmma<0><<<gemm_grid(ROWS, 2 * DI), 256, 0, stream>>>(
        normed, DM, in_proj_w + (size_t)layer * 2 * DI * DM, DM,
        nullptr, xz, ROWS, 2 * DI, DM);

    // causal depthwise conv + SiLU on x half
    mamba_conv_silu<<<(convN + 255) / 256, 256, 0, stream>>>(
        xz, conv_w + (size_t)layer * DI * DCONV, conv_b + (size_t)layer * DI,
        xconv);

    // x_proj: (ROWS, 80) = xconv (ROWS,DI) x W(80, DI)^T
    mamba_gemm_wmma<0><<<gemm_grid(ROWS, XDBL), 256, 0, stream>>>(
        xconv, DI, x_proj_w + (size_t)layer * XDBL * DI, DI,
        nullptr, xdbl, ROWS, XDBL, DI);

    // dt_proj + softplus: (ROWS, DI) = xdbl[:, :48] x W(DI, 48)^T + b
    mamba_gemm_wmma<1><<<gemm_grid(ROWS, DI), 256, 0, stream>>>(
        xdbl, XDBL, dt_proj_w + (size_t)layer * DI * DTRANK, DTRANK,
        dt_proj_b + (size_t)layer * DI, dtbuf, ROWS, DI, DTRANK);

    // selective scan (+ D skip + SiLU(z) gate), TDM-prefetched B/C
    mamba_scan<<<(B * DI) / 256, 256, 0, stream>>>(
        dtbuf, xconv, xdbl, xz,
        A_log + (size_t)layer * DI * DS, D_skip + (size_t)layer * DI, ybuf);

    // out_proj: hidden (ROWS, DM) = ybuf (ROWS,DI) x W(DM, DI)^T
    mamba_gemm_wmma<0><<<gemm_grid(ROWS, DM), 256, 0, stream>>>(
        ybuf, DI, out_proj_w + (size_t)layer * DM * DI, DI,
        nullptr, hidden, ROWS, DM, DI);
  }

  // final residual add + LayerNorm -> d_out
  mamba_add_ln<<<ROWS, 256, 0, stream>>>(
      hidden, residual, residual, (float*)d_out, norm_f_w, norm_f_b);
}